// ASGLoss_32779190403100
// MI455X (gfx1250) — compile-verified
//
#include <hip/hip_runtime.h>
#include <hip/hip_bf16.h>
#include <stdint.h>

#define T_DIM 2048
#define B_DIM 64
#define L_DIM 32
#define S_DIM 256
#define NEGINF (-1e30f)

typedef float v2f  __attribute__((ext_vector_type(2)));
typedef float v8f  __attribute__((ext_vector_type(8)));
typedef unsigned int u32x4 __attribute__((ext_vector_type(4)));
typedef int   i32x4 __attribute__((ext_vector_type(4)));
typedef int   i32x8 __attribute__((ext_vector_type(8)));

// ---------------------------------------------------------------------------
// Emission prefetch: 2KB contiguous slab global -> LDS.
// Primary path: CDNA5 Tensor Data Mover (tensor_load_to_lds + s_wait_tensorcnt).
// Fallback: plain vector loads (keeps compile green if builtin absent).
// ---------------------------------------------------------------------------
#if __has_builtin(__builtin_amdgcn_tensor_load_to_lds)
#define HAVE_TDM 1
__device__ __forceinline__ void tdm_fetch_2kb(const float* gsrc, unsigned lds_off) {
  unsigned long long ga = (unsigned long long)(uintptr_t)gsrc;
  u32x4 g0;
  g0[0] = 1u;                                                  // count=1, user mode
  g0[1] = lds_off;                                             // lds_addr (bytes)
  g0[2] = (unsigned)ga;                                        // global_addr[31:0]
  g0[3] = ((unsigned)(ga >> 32) & 0x01FFFFFFu) | (2u << 30);   // addr[56:32] | type=2
  i32x8 g1;
  g1[0] = (int)(2u << 16);      // workgroup_mask=0, data_size=2 (4 bytes)
  g1[1] = (int)(512u << 16);    // tensor_dim0 = 512 elements (bits 79:48, low part)
  g1[2] = (int)(1u << 16);      // tensor_dim0 hi = 0 | tensor_dim1 = 1
  g1[3] = (int)(512u << 16);    // tensor_dim1 hi = 0 | tile_dim0 = 512
  g1[4] = 1;                    // tile_dim1 = 1, tile_dim2 = 0
  g1[5] = 512;                  // tensor_dim0_stride = 512 (low 32)
  g1[6] = 0;                    // stride hi / tensor_dim1_stride lo
  g1[7] = 0;
  i32x4 gz = {0, 0, 0, 0};
#if defined(__clang_major__) && (__clang_major__ >= 23)
  i32x8 gz8 = {0, 0, 0, 0, 0, 0, 0, 0};
  __builtin_amdgcn_tensor_load_to_lds(g0, g1, gz, gz, gz8, 0);
#else
  __builtin_amdgcn_tensor_load_to_lds(g0, g1, gz, gz, 0);
#endif
}
#endif

__device__ __forceinline__ void fetch_emit(const float* gsrc, float* ldst, int lane) {
#if defined(HAVE_TDM)
  (void)lane;
  tdm_fetch_2kb(gsrc, (unsigned)(uintptr_t)ldst);   // generic-LDS low 32 bits = LDS offset
#else
  #pragma unroll
  for (int k = 0; k < 4; ++k)
    ((float4*)ldst)[k * 32 + lane] = ((const float4*)gsrc)[k * 32 + lane];
#endif
}

__device__ __forceinline__ void wait_emit(bool allow_one_pending) {
#if defined(HAVE_TDM)
  if (allow_one_pending) __builtin_amdgcn_s_wait_tensorcnt(1);
  else                   __builtin_amdgcn_s_wait_tensorcnt(0);
#endif
  asm volatile("" ::: "memory");
}

// ---------------------------------------------------------------------------
// FCC kernel: one wave (32 lanes) per 16-sequence tile; log-semiring matvec
// per step done as WMMA f32 16x16x4 against E = exp(transition) held in VGPRs.
// ---------------------------------------------------------------------------
__global__ __launch_bounds__(32) void asg_fcc_kernel(
    const float* __restrict__ trans,    // [32][32]
    const float* __restrict__ inputs,   // [T][B][L]
    float* __restrict__ fcc_out)        // [B]
{
  __shared__ float lds_alpha[L_DIM][16];   // alpha[j][b]
  __shared__ float lds_m[16];              // per-batch running max
  __shared__ float lds_emit[2][512];       // double-buffered emissions [b][l]

  const int lane  = threadIdx.x;
  const int b0    = blockIdx.x * 16;
  const int mrow  = lane & 15;             // N / M row within half
  const int khalf = lane >> 4;             // lane half (selects K/M offset)

  // A fragments for E[i][j] = exp(trans[i][j]); A 16x4 layout:
  // lane l (half h): VGPR0 -> K = 2h, VGPR1 -> K = 2h+1, M = l&15.
  v2f Afrag[2][8];
#pragma unroll
  for (int mt = 0; mt < 2; ++mt)
#pragma unroll
    for (int kt = 0; kt < 8; ++kt) {
      const int i = mt * 16 + mrow;
      Afrag[mt][kt].x = __expf(trans[i * L_DIM + kt * 4 + 2 * khalf + 0]);
      Afrag[mt][kt].y = __expf(trans[i * L_DIM + kt * 4 + 2 * khalf + 1]);
    }

  // alpha_0 = inputs[0, b0:b0+16, :]  (512 contiguous floats)
#pragma unroll
  for (int k = 0; k < 16; ++k) {
    const int idx = k * 32 + lane;
    lds_alpha[idx & 31][idx >> 5] = inputs[(size_t)b0 * L_DIM + idx];
  }
  __builtin_amdgcn_wave_barrier();

  // prefetch emissions for t = 1
  fetch_emit(inputs + (size_t)1 * B_DIM * L_DIM + (size_t)b0 * L_DIM,
             &lds_emit[1][0], lane);

  for (int t = 1; t < T_DIM; ++t) {
    if (t + 1 < T_DIM)
      fetch_emit(inputs + (size_t)(t + 1) * B_DIM * L_DIM + (size_t)b0 * L_DIM,
                 &lds_emit[(t + 1) & 1][0], lane);

    // per-batch max of alpha (for scaled-exp logsumexp)
    if (lane < 16) {
      float m = NEGINF;
#pragma unroll
      for (int j = 0; j < L_DIM; ++j) m = fmaxf(m, lds_alpha[j][lane]);
      lds_m[lane] = m;
    }
    __builtin_amdgcn_wave_barrier();
    const float mb = lds_m[mrow];

    // S[i][b] = sum_j E[i][j] * exp(alpha[j][b] - m[b])  via 16x WMMA f32 16x16x4
    v8f C0 = {0.f, 0.f, 0.f, 0.f, 0.f, 0.f, 0.f, 0.f};
    v8f C1 = {0.f, 0.f, 0.f, 0.f, 0.f, 0.f, 0.f, 0.f};
#pragma unroll
    for (int kt = 0; kt < 8; ++kt) {
      v2f Bf;   // B 4x16 layout: VGPR0 -> K = 2h, VGPR1 -> K = 2h+1, N = lane&15
      Bf.x = __expf(lds_alpha[kt * 4 + 2 * khalf + 0][mrow] - mb);
      Bf.y = __expf(lds_alpha[kt * 4 + 2 * khalf + 1][mrow] - mb);
      C0 = __builtin_amdgcn_wmma_f32_16x16x4_f32(false, Afrag[0][kt], false, Bf,
                                                 (short)0, C0, false, false);
      C1 = __builtin_amdgcn_wmma_f32_16x16x4_f32(false, Afrag[1][kt], false, Bf,
                                                 (short)0, C1, false, false);
    }

    wait_emit(t + 1 < T_DIM);
    const float* eb = &lds_emit[t & 1][0];   // eb[b*32 + l]

    // alpha_new[i][b] = emit + m[b] + log(S[i][b]);  C layout: VGPR r -> M = r + 8h
#pragma unroll
    for (int r = 0; r < 8; ++r) {
      const int i0 = r + 8 * khalf;
      const int i1 = 16 + i0;
      lds_alpha[i0][mrow] = eb[mrow * 32 + i0] + mb + __logf(C0[r]);
      lds_alpha[i1][mrow] = eb[mrow * 32 + i1] + mb + __logf(C1[r]);
    }
    __builtin_amdgcn_wave_barrier();
  }

  // fcc[b] = logsumexp_i alpha[T-1][i][b]
  if (lane < 16) {
    float m = NEGINF;
#pragma unroll
    for (int j = 0; j < L_DIM; ++j) m = fmaxf(m, lds_alpha[j][lane]);
    float s = 0.0f;
#pragma unroll
    for (int j = 0; j < L_DIM; ++j) s += __expf(lds_alpha[j][lane] - m);
    fcc_out[b0 + lane] = m + __logf(s);
  }
}

// ---------------------------------------------------------------------------
// FAC kernel: one 256-thread block per sequence; thread s owns target slot s.
// ---------------------------------------------------------------------------
__global__ __launch_bounds__(S_DIM) void asg_fac_kernel(
    const float* __restrict__ trans,
    const float* __restrict__ inputs,
    const int* __restrict__ targets,    // [B][S]
    float* __restrict__ fac_out)        // [B]
{
  const int b = blockIdx.x;
  const int s = threadIdx.x;

  __shared__ float sh_alpha[S_DIM];
  __shared__ float sh_emit[L_DIM];

  const int tgt  = targets[b * S_DIM + s];
  const int prv  = (s == 0) ? tgt : targets[b * S_DIM + s - 1];
  const float self_tr = trans[tgt * L_DIM + tgt];
  const float move_tr = trans[tgt * L_DIM + prv];

  float alpha = (s == 0) ? inputs[b * L_DIM + tgt] : NEGINF;

  for (int t = 1; t < T_DIM; ++t) {
    sh_alpha[s] = alpha;
    if (s < L_DIM) sh_emit[s] = inputs[(size_t)t * B_DIM * L_DIM + b * L_DIM + s];
    __syncthreads();
    const float e     = sh_emit[tgt];
    const float stay  = alpha + self_tr;
    const float moved = (s == 0) ? NEGINF : (sh_alpha[s - 1] + move_tr);
    const float mx = fmaxf(stay, moved);
    const float mn = fminf(stay, moved);
    alpha = e + mx + __logf(1.0f + __expf(mn - mx));
    __syncthreads();
  }
  if (s == S_DIM - 1) fac_out[b] = alpha;
}

// ---------------------------------------------------------------------------
// Finalize: mean(fcc - fac)
// ---------------------------------------------------------------------------
__global__ __launch_bounds__(B_DIM) void asg_finalize(
    const float* __restrict__ fcc, const float* __restrict__ fac,
    float* __restrict__ out)
{
  __shared__ float sh[B_DIM];
  const int i = threadIdx.x;
  sh[i] = fcc[i] - fac[i];
  __syncthreads();
  if (i == 0) {
    float s = 0.0f;
    for (int j = 0; j < B_DIM; ++j) s += sh[j];
    out[0] = s / (float)B_DIM;
  }
}

extern "C" void kernel_launch(void* const* d_in, const int* in_sizes, int n_in,
                              void* d_out, int out_size, void* d_ws, size_t ws_size,
                              hipStream_t stream) {
  (void)in_sizes; (void)n_in; (void)out_size; (void)ws_size;
  const float* trans   = (const float*)d_in[0];  // [L,L]
  const float* inputs  = (const float*)d_in[1];  // [T,B,L]
  const int*   targets = (const int*)d_in[2];    // [B,S]
  // d_in[3]=input_lengths (all T), d_in[4]=target_lengths (all S): compile-time here.

  float* ws_fcc = (float*)d_ws;
  float* ws_fac = ws_fcc + B_DIM;

  asg_fcc_kernel<<<B_DIM / 16, 32, 0, stream>>>(trans, inputs, ws_fcc);
  asg_fac_kernel<<<B_DIM, S_DIM, 0, stream>>>(trans, inputs, targets, ws_fac);
  asg_finalize<<<1, B_DIM, 0, stream>>>(ws_fcc, ws_fac, (float*)d_out);
}